// HardestContrastiveLoss_2173253452344
// MI455X (gfx1250) — compile-verified
//
#include <hip/hip_runtime.h>
#include <hip/hip_bf16.h>

#define NPOS   4096
#define NCAND  8192
#define DIM    64
#define NPAIRS 10000
#define NSPLIT 8
#define CHUNK  (NCAND / NSPLIT)   // 1024 columns per partial chunk

typedef float v2f __attribute__((ext_vector_type(2)));
typedef float v8f __attribute__((ext_vector_type(8)));

// ---------------------------------------------------------------------------
// Kernel 1: gather rows + squared norms + per-row positive-loss term
// One wave (32 lanes) per row; each lane handles 2 of the 64 dims.
// ---------------------------------------------------------------------------
__global__ void __launch_bounds__(256)
gather_norm_kernel(const float* __restrict__ feats0, const float* __restrict__ feats1,
                   const int* __restrict__ pos_pairs, const int* __restrict__ sel,
                   const int* __restrict__ cand0, const int* __restrict__ cand1,
                   float* __restrict__ posA0, float* __restrict__ posA1,
                   float* __restrict__ candB0, float* __restrict__ candB1,
                   float* __restrict__ na0, float* __restrict__ na1,
                   float* __restrict__ nb0, float* __restrict__ nb1,
                   int* __restrict__ posIdx0, int* __restrict__ posIdx1,
                   float* __restrict__ posTerm)
{
    int wave = (blockIdx.x * blockDim.x + threadIdx.x) >> 5;
    int lane = threadIdx.x & 31;

    if (wave < NPOS) {
        int i  = wave;
        int p  = sel[i];
        int i0 = pos_pairs[2 * p + 0];
        int i1 = pos_pairs[2 * p + 1];
        v2f f0 = *(const v2f*)(feats0 + (size_t)i0 * DIM + 2 * lane);
        v2f f1 = *(const v2f*)(feats1 + (size_t)i1 * DIM + 2 * lane);
        *(v2f*)(posA0 + (size_t)i * DIM + 2 * lane) = f0;
        *(v2f*)(posA1 + (size_t)i * DIM + 2 * lane) = f1;
        float n0 = f0.x * f0.x + f0.y * f0.y;
        float n1 = f1.x * f1.x + f1.y * f1.y;
        float dx = f0.x - f1.x, dy = f0.y - f1.y;
        float dd = dx * dx + dy * dy;
        #pragma unroll
        for (int m = 16; m >= 1; m >>= 1) {
            n0 += __shfl_xor(n0, m, 32);
            n1 += __shfl_xor(n1, m, 32);
            dd += __shfl_xor(dd, m, 32);
        }
        if (lane == 0) {
            na0[i] = n0;
            na1[i] = n1;
            posIdx0[i] = i0;
            posIdx1[i] = i1;
            float t = fmaxf(sqrtf(dd) - 0.1f, 0.0f);
            posTerm[i] = t * t;
        }
    } else if (wave < NPOS + NCAND) {
        int j = wave - NPOS;
        int c = cand0[j];
        v2f f = *(const v2f*)(feats0 + (size_t)c * DIM + 2 * lane);
        *(v2f*)(candB0 + (size_t)j * DIM + 2 * lane) = f;
        float n = f.x * f.x + f.y * f.y;
        #pragma unroll
        for (int m = 16; m >= 1; m >>= 1) n += __shfl_xor(n, m, 32);
        if (lane == 0) nb0[j] = n;
    } else if (wave < NPOS + 2 * NCAND) {
        int j = wave - NPOS - NCAND;
        int c = cand1[j];
        v2f f = *(const v2f*)(feats1 + (size_t)c * DIM + 2 * lane);
        *(v2f*)(candB1 + (size_t)j * DIM + 2 * lane) = f;
        float n = f.x * f.x + f.y * f.y;
        #pragma unroll
        for (int m = 16; m >= 1; m >>= 1) n += __shfl_xor(n, m, 32);
        if (lane == 0) nb1[j] = n;
    }
}

// ---------------------------------------------------------------------------
// Kernel 2: fused f32 WMMA GEMM + partial row-wise min/argmin.
// Grid: (NPOS/128, 2 sides, NSPLIT chunks). Block = 256 threads = 8 waves;
// wave w owns rows [blk.x*128 + 16w, +16); block scans its 1024-column chunk
// in 32-column super-tiles staged in LDS. Two independent WMMA accumulator
// chains per iteration (two 16x16 N-tiles) keep the matrix pipe busy.
// Tracks min over j of (||b_j||^2 - 2 a_i.b_j); na added in finalize.
// ---------------------------------------------------------------------------
__global__ void __launch_bounds__(256)
wmma_min_kernel(const float* __restrict__ posA0, const float* __restrict__ posA1,
                const float* __restrict__ candB0, const float* __restrict__ candB1,
                const float* __restrict__ nb0, const float* __restrict__ nb1,
                float* __restrict__ pMin0, int* __restrict__ pIdx0,
                float* __restrict__ pMin1, int* __restrict__ pIdx1)
{
    const float* A; const float* B; const float* nb;
    float* outMin; int* outIdx;
    if (blockIdx.y == 0) { A = posA0; B = candB1; nb = nb1; outMin = pMin0; outIdx = pIdx0; }
    else                 { A = posA1; B = candB0; nb = nb0; outMin = pMin1; outIdx = pIdx1; }

    __shared__ float bs[32 * 66];  // 32 cols x 64 K, padded stride 66 (conflict-free)

    int tid  = threadIdx.x;
    int wave = tid >> 5;
    int lane = tid & 31;
    int h    = lane >> 4;   // half-wave: selects K pair (A) / K row pair (B)
    int m    = lane & 15;   // row (A operand) / col (B operand, C col)
    int rowBase  = blockIdx.x * 128 + wave * 16;
    int chunk    = blockIdx.z;
    int colStart = chunk * CHUNK;

    // A operand resident in registers: lane needs A[rowBase+m][4t + 2h + {0,1}]
    v2f a[16];
    const float* arow = A + (size_t)(rowBase + m) * DIM + 2 * h;
    #pragma unroll
    for (int t = 0; t < 16; ++t) a[t] = *(const v2f*)(arow + 4 * t);

    float best[8];
    int   bidx[8];
    #pragma unroll
    for (int r = 0; r < 8; ++r) { best[r] = 3.0e38f; bidx[r] = 0; }

    for (int colBase = colStart; colBase < colStart + CHUNK; colBase += 32) {
        // cooperative 32x64 B-tile load: 2048 floats, 8 per thread
        {
            int c = tid >> 3;          // 0..31 column within super-tile
            int k = (tid & 7) * 8;     // 0..56
            const float* src = B + (size_t)(colBase + c) * DIM + k;
            v2f u0 = *(const v2f*)(src + 0);
            v2f u1 = *(const v2f*)(src + 2);
            v2f u2 = *(const v2f*)(src + 4);
            v2f u3 = *(const v2f*)(src + 6);
            float* dst = &bs[c * 66 + k];
            *(v2f*)(dst + 0) = u0;
            *(v2f*)(dst + 2) = u1;
            *(v2f*)(dst + 4) = u2;
            *(v2f*)(dst + 6) = u3;
        }
        __syncthreads();

        v8f acc0 = {};
        v8f acc1 = {};
        #pragma unroll
        for (int t = 0; t < 16; ++t) {
            v2f b0 = *(const v2f*)&bs[(m +  0) * 66 + 4 * t + 2 * h];
            v2f b1 = *(const v2f*)&bs[(m + 16) * 66 + 4 * t + 2 * h];
            acc0 = __builtin_amdgcn_wmma_f32_16x16x4_f32(
                       false, a[t], false, b0, (short)0, acc0, false, false);
            acc1 = __builtin_amdgcn_wmma_f32_16x16x4_f32(
                       false, a[t], false, b1, (short)0, acc1, false, false);
        }

        float nbv0 = nb[colBase + m];
        float nbv1 = nb[colBase + 16 + m];
        int   c0   = colBase + m;
        int   c1   = colBase + 16 + m;
        #pragma unroll
        for (int r = 0; r < 8; ++r) {
            float v0 = nbv0 - 2.0f * acc0[r];
            if (v0 < best[r]) { best[r] = v0; bidx[r] = c0; }  // ascending scan: < keeps first
            float v1 = nbv1 - 2.0f * acc1[r];
            if (v1 < best[r]) { best[r] = v1; bidx[r] = c1; }
        }
        __syncthreads();
    }

    // Reduce across the 16 lanes of each half-wave (C layout: col = lane%16,
    // row = r + 8*(lane/16)). Tie-break toward smaller column index.
    #pragma unroll
    for (int r = 0; r < 8; ++r) {
        #pragma unroll
        for (int msk = 8; msk >= 1; msk >>= 1) {
            float ov = __shfl_xor(best[r], msk, 32);
            int   oi = __shfl_xor(bidx[r], msk, 32);
            if (ov < best[r] || (ov == best[r] && oi < bidx[r])) {
                best[r] = ov; bidx[r] = oi;
            }
        }
    }
    if (m == 0) {
        int row = rowBase + 8 * h;
        #pragma unroll
        for (int r = 0; r < 8; ++r) {
            outMin[chunk * NPOS + row + r] = best[r];
            outIdx[chunk * NPOS + row + r] = bidx[r];
        }
    }
}

// ---------------------------------------------------------------------------
// Kernel 3: per-row finalize: combine chunk partials (ascending chunk order
// preserves first-index tie-break), distance, NN index mapping, pair-
// membership mask (pairs are L2-cached broadcast reads), masked neg terms.
// ---------------------------------------------------------------------------
__global__ void __launch_bounds__(256)
finalize_kernel(const int* __restrict__ pos_pairs,
                const int* __restrict__ cand0, const int* __restrict__ cand1,
                const float* __restrict__ na0, const float* __restrict__ na1,
                const float* __restrict__ pMin0, const int* __restrict__ pIdx0,
                const float* __restrict__ pMin1, const int* __restrict__ pIdx1,
                const int* __restrict__ posIdx0, const int* __restrict__ posIdx1,
                float* __restrict__ negTerm0, float* __restrict__ cnt0,
                float* __restrict__ negTerm1, float* __restrict__ cnt1)
{
    int gid = blockIdx.x * blockDim.x + threadIdx.x;
    if (gid >= 2 * NPOS) return;
    int side = gid >> 12;        // NPOS == 4096 == 2^12
    int i    = gid & (NPOS - 1);

    const float* pMin = (side == 0) ? pMin0 : pMin1;
    const int*   pIdx = (side == 0) ? pIdx0 : pIdx1;

    float bestV = 3.0e38f;
    int   bestI = 0;
    #pragma unroll
    for (int c = 0; c < NSPLIT; ++c) {
        float v  = pMin[c * NPOS + i];
        int   ix = pIdx[c * NPOS + i];
        if (v < bestV || (v == bestV && ix < bestI)) { bestV = v; bestI = ix; }
    }

    int a, b; float d2;
    if (side == 0) {
        d2 = na0[i] + bestV;
        a  = posIdx0[i];
        b  = cand1[bestI];
    } else {
        d2 = na1[i] + bestV;
        a  = cand0[bestI];
        b  = posIdx1[i];
    }
    float d = sqrtf(fmaxf(d2, 0.0f));

    const int2* pp = (const int2*)pos_pairs;
    bool inPairs = false;
    for (int p = 0; p < NPAIRS; ++p) {
        int2 q = pp[p];
        inPairs = inPairs || (q.x == a && q.y == b);
    }
    float t    = fmaxf(1.4f - d, 0.0f);
    float term = inPairs ? 0.0f : t * t;
    float cnt  = inPairs ? 0.0f : 1.0f;
    if (side == 0) { negTerm0[i] = term; cnt0[i] = cnt; }
    else           { negTerm1[i] = term; cnt1[i] = cnt; }
}

// ---------------------------------------------------------------------------
// Kernel 4: single-block deterministic reduction -> (total, pos, neg)
// ---------------------------------------------------------------------------
__global__ void __launch_bounds__(256)
reduce_kernel(const float* __restrict__ posTerm,
              const float* __restrict__ negTerm0, const float* __restrict__ cnt0,
              const float* __restrict__ negTerm1, const float* __restrict__ cnt1,
              float* __restrict__ out)
{
    __shared__ float s[5][256];
    int tid = threadIdx.x;
    float sp = 0.f, sn0 = 0.f, sc0 = 0.f, sn1 = 0.f, sc1 = 0.f;
    for (int i = tid; i < NPOS; i += 256) {
        sp  += posTerm[i];
        sn0 += negTerm0[i];
        sc0 += cnt0[i];
        sn1 += negTerm1[i];
        sc1 += cnt1[i];
    }
    s[0][tid] = sp; s[1][tid] = sn0; s[2][tid] = sc0; s[3][tid] = sn1; s[4][tid] = sc1;
    __syncthreads();
    for (int off = 128; off > 0; off >>= 1) {
        if (tid < off) {
            #pragma unroll
            for (int q = 0; q < 5; ++q) s[q][tid] += s[q][tid + off];
        }
        __syncthreads();
    }
    if (tid == 0) {
        float pos_loss = s[0][0] / (float)NPOS;
        float nl0 = s[1][0] / fmaxf(s[2][0], 1.0f);
        float nl1 = s[3][0] / fmaxf(s[4][0], 1.0f);
        float neg = 0.5f * (nl0 + nl1);
        out[0] = pos_loss + neg;
        out[1] = pos_loss;
        out[2] = neg;
    }
}

// ---------------------------------------------------------------------------
extern "C" void kernel_launch(void* const* d_in, const int* in_sizes, int n_in,
                              void* d_out, int out_size, void* d_ws, size_t ws_size,
                              hipStream_t stream)
{
    (void)in_sizes; (void)n_in; (void)out_size; (void)ws_size;

    const float* feats0 = (const float*)d_in[0];
    const float* feats1 = (const float*)d_in[1];
    const int*   pairs  = (const int*)d_in[2];
    const int*   sel    = (const int*)d_in[3];
    const int*   cand0  = (const int*)d_in[4];
    const int*   cand1  = (const int*)d_in[5];
    float*       out    = (float*)d_out;

    // workspace carve-up (floats / ints, all 4-byte elements)
    float* w = (float*)d_ws;
    size_t off = 0;
    float* posA0   = w + off; off += (size_t)NPOS * DIM;
    float* posA1   = w + off; off += (size_t)NPOS * DIM;
    float* candB0  = w + off; off += (size_t)NCAND * DIM;
    float* candB1  = w + off; off += (size_t)NCAND * DIM;
    float* na0     = w + off; off += NPOS;
    float* na1     = w + off; off += NPOS;
    float* nb0     = w + off; off += NCAND;
    float* nb1     = w + off; off += NCAND;
    float* pMin0   = w + off; off += (size_t)NSPLIT * NPOS;
    float* pMin1   = w + off; off += (size_t)NSPLIT * NPOS;
    int*   pIdx0   = (int*)(w + off); off += (size_t)NSPLIT * NPOS;
    int*   pIdx1   = (int*)(w + off); off += (size_t)NSPLIT * NPOS;
    int*   posIdx0 = (int*)(w + off); off += NPOS;
    int*   posIdx1 = (int*)(w + off); off += NPOS;
    float* posTerm = w + off; off += NPOS;
    float* negT0   = w + off; off += NPOS;
    float* cnt0    = w + off; off += NPOS;
    float* negT1   = w + off; off += NPOS;
    float* cnt1    = w + off; off += NPOS;

    // 1) gather + norms: one wave per row, 8 waves per block
    {
        int waves  = NPOS + 2 * NCAND;          // 20480
        int blocks = waves / 8;                 // 2560
        gather_norm_kernel<<<blocks, 256, 0, stream>>>(
            feats0, feats1, pairs, sel, cand0, cand1,
            posA0, posA1, candB0, candB1,
            na0, na1, nb0, nb1, posIdx0, posIdx1, posTerm);
    }

    // 2) fused WMMA GEMM + partial row min/argmin, both directions, 8 N-chunks
    {
        dim3 grid(NPOS / 128, 2, NSPLIT);       // 32 x 2 x 8 = 512 blocks
        wmma_min_kernel<<<grid, 256, 0, stream>>>(
            posA0, posA1, candB0, candB1, nb0, nb1,
            pMin0, pIdx0, pMin1, pIdx1);
    }

    // 3) per-row finalize (both sides), combines the 8 chunk partials
    finalize_kernel<<<(2 * NPOS) / 256, 256, 0, stream>>>(
        pairs, cand0, cand1, na0, na1,
        pMin0, pIdx0, pMin1, pIdx1, posIdx0, posIdx1,
        negT0, cnt0, negT1, cnt1);

    // 4) deterministic final reduction
    reduce_kernel<<<1, 256, 0, stream>>>(posTerm, negT0, cnt0, negT1, cnt1, out);
}